// BaconAdditionReasoner_52192442581023
// MI455X (gfx1250) — compile-verified
//
#include <hip/hip_runtime.h>
#include <hip/hip_bf16.h>

typedef __attribute__((ext_vector_type(2))) float v2f;
typedef __attribute__((ext_vector_type(8))) float v8f;

#define LSTRIDE 17        // l1s/l2s row stride (floats); 17 coprime 64 banks
#define PSTRIDE 12        // staged p1/p2 row stride (floats); K padded 10->12

// LDS layout (floats):
//   [0      , 3072 )  stageP1  256 rows * 12
//   [3072   , 6144 )  stageP2  256 rows * 12
//   [6144   , 10496)  l1s      256 rows * 17
//   [10496  , 14848)  l2s      256 rows * 17
// stageP1+stageP2 region (6144 floats) is reused for the 256*19 output stage.
#define SMEM_FLOATS 14848
#define OFF_P1 0
#define OFF_P2 3072
#define OFF_L1 6144
#define OFF_L2 10496
#define OFF_Y  0          // reuse of stage region after phase 1

// ---------------------------------------------------------------------------
// Prologue: soft-permutation matrices S1,S2 (10x10) stored transposed and
// zero-padded to [12][16] each: ST[m][k][n] = S_m[n][k]. Every entry computed
// independently -> race-free, no barriers.
// ---------------------------------------------------------------------------
__global__ void bacon_soft_perm_prep(const float* __restrict__ W1,
                                     const float* __restrict__ W2,
                                     float* __restrict__ ST) {
    for (int i = threadIdx.x; i < 2 * 12 * 16; i += blockDim.x) {
        int mm  = i / 192;
        int rem = i - mm * 192;
        int k   = rem >> 4;         // 0..11
        int n   = rem & 15;         // 0..15
        float val = 0.0f;
        if (k < 10 && n < 10) {
            const float* W = mm ? W2 : W1;
            float lo = 3.4e38f, hi = -3.4e38f;
            #pragma unroll
            for (int j = 0; j < 10; ++j) {
                float w = W[n * 10 + j];
                lo = fminf(lo, w);
                hi = fmaxf(hi, w);
            }
            float inv = 1.0f / (hi - lo + 1e-8f);
            float sum = 0.0f;
            #pragma unroll
            for (int j = 0; j < 10; ++j) sum += (W[n * 10 + j] - lo) * inv;
            val = ((W[n * 10 + k] - lo) * inv) / (sum + 1e-8f);
        }
        ST[i] = val;
    }
}

// ---------------------------------------------------------------------------
// Fused main kernel. 256 threads = 8 waves, 256 batch rows per block.
// Phase 0: coalesced float2 staging of p1/p2 into LDS (stride 12, zero-pad).
// Phase 1: per wave, 2 groups of 16 rows; l1,l2 via V_WMMA_F32_16X16X4_F32
//          (K = 10 padded to 12 = 3 chunks of 4); A operands from LDS.
// Phase 2: one thread per row: 10x10 min/log, i+j buckets, exp, normalize.
// Phase 3: coalesced output store via LDS restage.
// ---------------------------------------------------------------------------
__global__ void __launch_bounds__(256)
bacon_addition_fused(const float* __restrict__ p1,
                     const float* __restrict__ p2,
                     const float* __restrict__ ST,   // [2][12][16]
                     float* __restrict__ out,        // [nB][19]
                     int nB) {
    __shared__ float smem[SMEM_FLOATS];

    const int t    = threadIdx.x;
    const int lane = t & 31;
    const int wave = t >> 5;
    const int hi   = lane >> 4;      // lane-half: K pair select (A/B), M+8 (C/D)
    const int ln   = lane & 15;      // M (A rows) / N (B,C,D cols)
    const int blockRow0 = blockIdx.x * 256;

    // ---- Phase 0: stage p1/p2 block chunk into LDS, padded to 12 floats/row
    {
        const float2* g1 = (const float2*)p1 + blockRow0 * 5;
        const float2* g2 = (const float2*)p2 + blockRow0 * 5;
        const int limit = nB * 5 - blockRow0 * 5;      // in-bounds float2 count
        #pragma unroll
        for (int it = 0; it < 5; ++it) {               // 1280 float2 / 256 thr
            int e  = t + it * 256;                     // float2 index in block
            int r  = e / 5;
            int c2 = e - r * 5;
            float2 z; z.x = 0.0f; z.y = 0.0f;
            float2 v1 = (e < limit) ? g1[e] : z;
            float2 v2 = (e < limit) ? g2[e] : z;
            *(float2*)&smem[OFF_P1 + r * PSTRIDE + 2 * c2] = v1;
            *(float2*)&smem[OFF_P2 + r * PSTRIDE + 2 * c2] = v2;
        }
        // zero K padding (cols 10,11) for this thread's row
        float2 z; z.x = 0.0f; z.y = 0.0f;
        *(float2*)&smem[OFF_P1 + t * PSTRIDE + 10] = z;
        *(float2*)&smem[OFF_P2 + t * PSTRIDE + 10] = z;
    }

    // B operands (4x16 K-major tiles of S^T), shared across both row groups.
    // VGPR0 = row (kx), VGPR1 = row (kx+1), lane = column N; kx = 4c + 2*hi.
    v2f b1[3], b2[3];
    #pragma unroll
    for (int c = 0; c < 3; ++c) {
        int kx = 4 * c + 2 * hi;                       // max 10; +1 = 11 < 12
        b1[c].x = ST[kx * 16 + ln];
        b1[c].y = ST[(kx + 1) * 16 + ln];
        b2[c].x = ST[192 + kx * 16 + ln];
        b2[c].y = ST[192 + (kx + 1) * 16 + ln];
    }

    __syncthreads();

    // ---- Phase 1: WMMA matvecs, results column-per-lane -> restage via LDS
    #pragma unroll
    for (int g = 0; g < 2; ++g) {
        const int lrow = wave * 32 + g * 16 + ln;      // local A row (M)
        v8f c1 = {};
        v8f c2 = {};
        #pragma unroll
        for (int c = 0; c < 3; ++c) {
            int kx = 4 * c + 2 * hi;                   // even -> 8B aligned
            float2 f1 = *(const float2*)&smem[OFF_P1 + lrow * PSTRIDE + kx];
            float2 f2 = *(const float2*)&smem[OFF_P2 + lrow * PSTRIDE + kx];
            v2f a1; a1.x = f1.x; a1.y = f1.y;
            v2f a2; a2.x = f2.x; a2.y = f2.y;
            c1 = __builtin_amdgcn_wmma_f32_16x16x4_f32(
                     false, a1, false, b1[c], (short)0, c1, false, false);
            c2 = __builtin_amdgcn_wmma_f32_16x16x4_f32(
                     false, a2, false, b2[c], (short)0, c2, false, false);
        }
        // C/D layout: VGPR v holds M = v + 8*hi, lane = column N
        const int lbase = wave * 32 + g * 16 + 8 * hi;
        #pragma unroll
        for (int v = 0; v < 8; ++v) {
            smem[OFF_L1 + (lbase + v) * LSTRIDE + ln] = c1[v];
            smem[OFF_L2 + (lbase + v) * LSTRIDE + ln] = c2[v];
        }
    }
    __syncthreads();

    // ---- Phase 2: one thread per batch row
    float a1v[10], a2v[10];
    #pragma unroll
    for (int i = 0; i < 10; ++i) {
        a1v[i] = smem[OFF_L1 + t * LSTRIDE + i];
        a2v[i] = smem[OFF_L2 + t * LSTRIDE + i];
    }

    float acc[19];
    #pragma unroll
    for (int k = 0; k < 19; ++k) acc[k] = 0.0f;

    #pragma unroll
    for (int i = 0; i < 10; ++i) {
        #pragma unroll
        for (int j = 0; j < 10; ++j) {
            float s = fminf(a1v[i], a2v[j]);
            s = fminf(fmaxf(s, 1e-6f), 1.0f - 1e-6f);
            acc[i + j] += __logf(1.0f - s + 1e-12f);
        }
    }

    float y[19];
    float sum = 0.0f;
    #pragma unroll
    for (int k = 0; k < 19; ++k) {
        y[k] = 1.0f - __expf(acc[k]);
        sum += y[k];
    }
    float inv = 1.0f / (sum + 1e-9f);

    // ---- Phase 3: restage y into (reused) LDS, coalesced store to global
    #pragma unroll
    for (int k = 0; k < 19; ++k) smem[OFF_Y + t * 19 + k] = y[k] * inv;
    __syncthreads();

    {
        float* gout = out + (size_t)blockRow0 * 19;
        const int limit = nB * 19 - blockRow0 * 19;    // in-bounds float count
        #pragma unroll
        for (int it = 0; it < 19; ++it) {              // 4864 / 256 = 19
            int e = t + it * 256;
            if (e < limit) gout[e] = smem[OFF_Y + e];
        }
    }
}

// ---------------------------------------------------------------------------
extern "C" void kernel_launch(void* const* d_in, const int* in_sizes, int n_in,
                              void* d_out, int out_size, void* d_ws, size_t ws_size,
                              hipStream_t stream) {
    const float* p1 = (const float*)d_in[0];
    const float* p2 = (const float*)d_in[1];
    const float* W1 = (const float*)d_in[2];
    const float* W2 = (const float*)d_in[3];
    float* out = (float*)d_out;
    float* ST  = (float*)d_ws;          // 2 * 12 * 16 floats = 1536 bytes

    const int nB = in_sizes[0] / 10;    // 262144

    bacon_soft_perm_prep<<<1, 64, 0, stream>>>(W1, W2, ST);

    const int grid = (nB + 255) / 256;
    bacon_addition_fused<<<grid, 256, 0, stream>>>(p1, p2, ST, out, nB);
}